// SSMLayer_25426206392924
// MI455X (gfx1250) — compile-verified
//
#include <hip/hip_runtime.h>
#include <math.h>

// ---------------------------------------------------------------------------
// SSM layer for MI455X (gfx1250, wave32):
//   g  = clip(sigmoid(x Wg^T), .05, .95)
//   a  = 1 - g (1 - A);  b = g (x B^T + bias_h)
//   h_t = a_t h_{t-1} + b_t        (chunked parallel scan)
//   y  = (h C^T + x D^T + bias_y) / sqrt(H)
// GEMMs: V_WMMA_F32_16X16X4_F32 with double-buffered LDS fed by
// GLOBAL_LOAD_ASYNC_TO_LDS_B128 (ASYNCcnt), overlapping DMA with WMMA.
// ---------------------------------------------------------------------------

typedef __attribute__((ext_vector_type(2))) float v2f;
typedef __attribute__((ext_vector_type(8))) float v8f;

#define DIM     512
#define BATCH   8
#define SEQ     4096
#define M_TOTAL (BATCH * SEQ)            // 32768 rows

#define NCH     (BATCH * DIM)            // 4096 scan channels
#define NCHUNK  16
#define CHUNK   (SEQ / NCHUNK)           // 256 steps per chunk

// ---- CDNA5 async global->LDS copy (16B per lane), with portable fallback ----
#if defined(__has_builtin)
#if __has_builtin(__builtin_amdgcn_global_load_async_to_lds_b128)
#define HAS_ASYNC_LDS 1
#endif
#endif
#ifndef HAS_ASYNC_LDS
#define HAS_ASYNC_LDS 0
#endif

#if HAS_ASYNC_LDS
#if __has_builtin(__builtin_amdgcn_s_wait_asynccnt)
#define WAIT_ASYNC(n) __builtin_amdgcn_s_wait_asynccnt(n)
#else
#define WAIT_ASYNC(n) asm volatile("s_wait_asynccnt %0" ::"n"(n) : "memory")
#endif
#else
#define WAIT_ASYNC(n) do { } while (0)
#endif

typedef int v4i_vs __attribute__((vector_size(16)));
typedef __attribute__((address_space(1))) v4i_vs* as1_v4i_p;
typedef __attribute__((address_space(3))) v4i_vs* as3_v4i_p;

__device__ __forceinline__ void cp16_async(const float* __restrict__ g,
                                           float* __restrict__ l) {
#if HAS_ASYNC_LDS
  __builtin_amdgcn_global_load_async_to_lds_b128(
      (as1_v4i_p)(void*)g, (as3_v4i_p)(void*)l,
      /*imm offset=*/0, /*cpol=*/0);
#else
  *(float4*)l = *(const float4*)g;
#endif
}

__device__ __forceinline__ v8f wmma_f32(v2f a, v2f b, v8f c) {
  // D = A(16x4, f32) * B(4x16, f32) + C(16x16, f32)
  return __builtin_amdgcn_wmma_f32_16x16x4_f32(
      /*neg_a=*/false, a, /*neg_b=*/false, b,
      /*c_mod=*/(short)0, c, /*reuse_a=*/false, /*reuse_b=*/false);
}

// Common GEMM tiling: 64(M) x 64(N) block tile, 256 threads = 8 wave32s,
// each wave owns a 16x32 strip. K chunked by 32 through double-buffered LDS.
#define MT 64
#define NT 64
#define KC 32
#define KP 36      // 36*4B = 144B row stride: 16B-aligned, conflict-free banks
#define KSTEPS (DIM / KC)   // 16 chunks

// ---------------------------------------------------------------------------
// Kernel 1: fused gate GEMMs. out[m,n] = sum_k X[m,k] * W[n,k], W in {Wg, B},
// epilogue emits a_coef / b_coef directly (g and xB never hit HBM).
// ---------------------------------------------------------------------------
__global__ __launch_bounds__(256) void gate_gemm_kernel(
    const float* __restrict__ X,  const float* __restrict__ Wg,
    const float* __restrict__ Bw, const float* __restrict__ Av,
    const float* __restrict__ bias_h,
    float* __restrict__ a_coef,   float* __restrict__ b_coef)
{
  __shared__ float xs[2][MT][KP];   // 3 arrays * 2 buffers * 9216B = 55296B LDS
  __shared__ float gs[2][NT][KP];
  __shared__ float bs[2][NT][KP];

  const int tid  = threadIdx.x;
  const int wave = tid >> 5;
  const int lane = tid & 31;
  const int m0 = blockIdx.y * MT;
  const int n0 = blockIdx.x * NT;
  const int wm = (wave >> 1) << 4;   // 0,16,32,48
  const int wn = (wave & 1) << 5;    // 0,32

  // per-thread copy slots: 64 rows x 32 cols per array, 16B per lane, 2 rounds
  const int r0 = tid >> 3;                 // rows for round 0
  const int c0 = (tid & 7) << 2;
  const int r1 = (tid + 256) >> 3;         // rows for round 1

  v8f ag0 = {}, ag1 = {}, ab0 = {}, ab1 = {};

  const int row = lane & 15;         // A: M index / B: N index
  const int khi = (lane >> 4) << 1;  // hi lane-half carries K+2,K+3

  // ---- prologue: stream chunk 0 into buffer 0 ----
  {
    const size_t gx0 = (size_t)(m0 + r0) * DIM + c0;
    const size_t gx1 = (size_t)(m0 + r1) * DIM + c0;
    const size_t gw0 = (size_t)(n0 + r0) * DIM + c0;
    const size_t gw1 = (size_t)(n0 + r1) * DIM + c0;
    cp16_async(X + gx0,  &xs[0][r0][c0]);  cp16_async(X + gx1,  &xs[0][r1][c0]);
    cp16_async(Wg + gw0, &gs[0][r0][c0]);  cp16_async(Wg + gw1, &gs[0][r1][c0]);
    cp16_async(Bw + gw0, &bs[0][r0][c0]);  cp16_async(Bw + gw1, &bs[0][r1][c0]);
  }

  for (int k = 0; k < KSTEPS; ++k) {
    const int cur = k & 1;
    if (k + 1 < KSTEPS) {              // stream next chunk into other buffer
      const int nxt = cur ^ 1;
      const int k0 = (k + 1) * KC;
      const size_t gx0 = (size_t)(m0 + r0) * DIM + k0 + c0;
      const size_t gx1 = (size_t)(m0 + r1) * DIM + k0 + c0;
      const size_t gw0 = (size_t)(n0 + r0) * DIM + k0 + c0;
      const size_t gw1 = (size_t)(n0 + r1) * DIM + k0 + c0;
      cp16_async(X + gx0,  &xs[nxt][r0][c0]);  cp16_async(X + gx1,  &xs[nxt][r1][c0]);
      cp16_async(Wg + gw0, &gs[nxt][r0][c0]);  cp16_async(Wg + gw1, &gs[nxt][r1][c0]);
      cp16_async(Bw + gw0, &bs[nxt][r0][c0]);  cp16_async(Bw + gw1, &bs[nxt][r1][c0]);
      WAIT_ASYNC(6);                   // 6 newer in flight -> current chunk landed
    } else {
      WAIT_ASYNC(0);
    }
    __syncthreads();                   // all waves' chunk-k data visible

    const float (*xc)[KP] = xs[cur];
    const float (*gc)[KP] = gs[cur];
    const float (*bc)[KP] = bs[cur];
#pragma unroll
    for (int k4 = 0; k4 < KC; k4 += 4) {
      const int kk = k4 + khi;
      v2f a;  a.x  = xc[wm + row][kk];      a.y  = xc[wm + row][kk + 1];
      v2f g0; g0.x = gc[wn + row][kk];      g0.y = gc[wn + row][kk + 1];
      v2f g1; g1.x = gc[wn + 16 + row][kk]; g1.y = gc[wn + 16 + row][kk + 1];
      v2f b0; b0.x = bc[wn + row][kk];      b0.y = bc[wn + row][kk + 1];
      v2f b1; b1.x = bc[wn + 16 + row][kk]; b1.y = bc[wn + 16 + row][kk + 1];
      ag0 = wmma_f32(a, g0, ag0);
      ag1 = wmma_f32(a, g1, ag1);
      ab0 = wmma_f32(a, b0, ab0);
      ab1 = wmma_f32(a, b1, ab1);
    }
    __syncthreads();                   // safe to overwrite buffer `cur`
  }

  // Epilogue: C/D layout -> VGPR r holds M=r (lanes 0-15) / M=r+8 (lanes 16-31)
  const int col  = lane & 15;
  const int rofs = (lane >> 4) << 3;
  const int h0 = n0 + wn + col;
  const int h1 = h0 + 16;
  const float A0 = Av[h0],      A1 = Av[h1];
  const float bh0 = bias_h[h0], bh1 = bias_h[h1];
#pragma unroll
  for (int r = 0; r < 8; ++r) {
    const int m = m0 + wm + rofs + r;
    const size_t o0 = (size_t)m * DIM + h0;
    const size_t o1 = (size_t)m * DIM + h1;
    float gg0 = 1.0f / (1.0f + __expf(-ag0[r]));
    gg0 = fminf(fmaxf(gg0, 0.05f), 0.95f);
    float gg1 = 1.0f / (1.0f + __expf(-ag1[r]));
    gg1 = fminf(fmaxf(gg1, 0.05f), 0.95f);
    a_coef[o0] = 1.0f - gg0 * (1.0f - A0);
    b_coef[o0] = gg0 * (ab0[r] + bh0);
    a_coef[o1] = 1.0f - gg1 * (1.0f - A1);
    b_coef[o1] = gg1 * (ab1[r] + bh1);
  }
}

// ---------------------------------------------------------------------------
// Chunked parallel scan: h_t = a_t h_{t-1} + b_t
// ---------------------------------------------------------------------------
__global__ __launch_bounds__(256) void scan_pass1(
    const float* __restrict__ a_coef, const float* __restrict__ b_coef,
    float* __restrict__ Pc, float* __restrict__ Qc)
{
  const int g = blockIdx.x * blockDim.x + threadIdx.x;  // 0..65535
  const int c = g & (NCH - 1);                          // channel (coalesced in h)
  const int chunk = g >> 12;
  const int b = c >> 9;
  const int h = c & (DIM - 1);
  size_t base = ((size_t)b * SEQ + (size_t)chunk * CHUNK) * DIM + h;
  float P = 1.0f, Q = 0.0f;
#pragma unroll 4
  for (int t = 0; t < CHUNK; ++t) {
    const float a = a_coef[base];
    Q = fmaf(a, Q, b_coef[base]);
    P *= a;
    base += DIM;
  }
  Pc[g] = P;
  Qc[g] = Q;
}

__global__ __launch_bounds__(256) void scan_pass2(
    const float* __restrict__ Pc, const float* __restrict__ Qc,
    float* __restrict__ Hst)
{
  const int c = blockIdx.x * blockDim.x + threadIdx.x;  // 0..4095
  float hrun = 0.0f;
#pragma unroll
  for (int j = 0; j < NCHUNK; ++j) {
    Hst[j * NCH + c] = hrun;                            // exclusive prefix
    hrun = fmaf(Pc[j * NCH + c], hrun, Qc[j * NCH + c]);
  }
}

__global__ __launch_bounds__(256) void scan_pass3(
    const float* __restrict__ a_coef, const float* __restrict__ b_coef,
    const float* __restrict__ Hst, float* __restrict__ hs)
{
  const int g = blockIdx.x * blockDim.x + threadIdx.x;
  const int c = g & (NCH - 1);
  const int chunk = g >> 12;
  const int b = c >> 9;
  const int h = c & (DIM - 1);
  size_t base = ((size_t)b * SEQ + (size_t)chunk * CHUNK) * DIM + h;
  float hv = Hst[g];
#pragma unroll 4
  for (int t = 0; t < CHUNK; ++t) {
    hv = fmaf(a_coef[base], hv, b_coef[base]);
    hs[base] = hv;
    base += DIM;
  }
}

// ---------------------------------------------------------------------------
// Kernel 5: fused output GEMMs. y[m,o] = (sum_h hs[m,h]C[o,h] +
//            sum_i x[m,i]D[o,i] + bias_y[o]) / sqrt(H)
// ---------------------------------------------------------------------------
__global__ __launch_bounds__(256) void out_gemm_kernel(
    const float* __restrict__ HS, const float* __restrict__ X,
    const float* __restrict__ Cw, const float* __restrict__ Dw,
    const float* __restrict__ bias_y, float* __restrict__ Y)
{
  __shared__ float hss[2][MT][KP];  // 4 arrays * 2 buffers * 9216B = 73728B LDS
  __shared__ float xs [2][MT][KP];
  __shared__ float cs [2][NT][KP];
  __shared__ float ds [2][NT][KP];

  const int tid  = threadIdx.x;
  const int wave = tid >> 5;
  const int lane = tid & 31;
  const int m0 = blockIdx.y * MT;
  const int n0 = blockIdx.x * NT;
  const int wm = (wave >> 1) << 4;
  const int wn = (wave & 1) << 5;

  const int r0 = tid >> 3;
  const int c0 = (tid & 7) << 2;
  const int r1 = (tid + 256) >> 3;

  v8f acc0 = {}, acc1 = {};

  const int row = lane & 15;
  const int khi = (lane >> 4) << 1;

  {
    const size_t gm0 = (size_t)(m0 + r0) * DIM + c0;
    const size_t gm1 = (size_t)(m0 + r1) * DIM + c0;
    const size_t gn0 = (size_t)(n0 + r0) * DIM + c0;
    const size_t gn1 = (size_t)(n0 + r1) * DIM + c0;
    cp16_async(HS + gm0, &hss[0][r0][c0]);  cp16_async(HS + gm1, &hss[0][r1][c0]);
    cp16_async(X  + gm0, &xs [0][r0][c0]);  cp16_async(X  + gm1, &xs [0][r1][c0]);
    cp16_async(Cw + gn0, &cs [0][r0][c0]);  cp16_async(Cw + gn1, &cs [0][r1][c0]);
    cp16_async(Dw + gn0, &ds [0][r0][c0]);  cp16_async(Dw + gn1, &ds [0][r1][c0]);
  }

  for (int k = 0; k < KSTEPS; ++k) {
    const int cur = k & 1;
    if (k + 1 < KSTEPS) {
      const int nxt = cur ^ 1;
      const int k0 = (k + 1) * KC;
      const size_t gm0 = (size_t)(m0 + r0) * DIM + k0 + c0;
      const size_t gm1 = (size_t)(m0 + r1) * DIM + k0 + c0;
      const size_t gn0 = (size_t)(n0 + r0) * DIM + k0 + c0;
      const size_t gn1 = (size_t)(n0 + r1) * DIM + k0 + c0;
      cp16_async(HS + gm0, &hss[nxt][r0][c0]);  cp16_async(HS + gm1, &hss[nxt][r1][c0]);
      cp16_async(X  + gm0, &xs [nxt][r0][c0]);  cp16_async(X  + gm1, &xs [nxt][r1][c0]);
      cp16_async(Cw + gn0, &cs [nxt][r0][c0]);  cp16_async(Cw + gn1, &cs [nxt][r1][c0]);
      cp16_async(Dw + gn0, &ds [nxt][r0][c0]);  cp16_async(Dw + gn1, &ds [nxt][r1][c0]);
      WAIT_ASYNC(8);
    } else {
      WAIT_ASYNC(0);
    }
    __syncthreads();

    const float (*hc)[KP] = hss[cur];
    const float (*xc)[KP] = xs[cur];
    const float (*cc)[KP] = cs[cur];
    const float (*dc)[KP] = ds[cur];
#pragma unroll
    for (int k4 = 0; k4 < KC; k4 += 4) {
      const int kk = k4 + khi;
      v2f ah; ah.x = hc[wm + row][kk];      ah.y = hc[wm + row][kk + 1];
      v2f ax; ax.x = xc[wm + row][kk];      ax.y = xc[wm + row][kk + 1];
      v2f w0; w0.x = cc[wn + row][kk];      w0.y = cc[wn + row][kk + 1];
      v2f w1; w1.x = cc[wn + 16 + row][kk]; w1.y = cc[wn + 16 + row][kk + 1];
      v2f d0; d0.x = dc[wn + row][kk];      d0.y = dc[wn + row][kk + 1];
      v2f d1; d1.x = dc[wn + 16 + row][kk]; d1.y = dc[wn + 16 + row][kk + 1];
      acc0 = wmma_f32(ah, w0, acc0);
      acc0 = wmma_f32(ax, d0, acc0);
      acc1 = wmma_f32(ah, w1, acc1);
      acc1 = wmma_f32(ax, d1, acc1);
    }
    __syncthreads();
  }

  const float scale = 0.04419417382415922f;  // 1/sqrt(512)
  const int col  = lane & 15;
  const int rofs = (lane >> 4) << 3;
  const int o0 = n0 + wn + col;
  const int o1 = o0 + 16;
  const float by0 = bias_y[o0], by1 = bias_y[o1];
#pragma unroll
  for (int r = 0; r < 8; ++r) {
    const int m = m0 + wm + rofs + r;
    Y[(size_t)m * DIM + o0] = (acc0[r] + by0) * scale;
    Y[(size_t)m * DIM + o1] = (acc1[r] + by1) * scale;
  }
}

// ---------------------------------------------------------------------------
extern "C" void kernel_launch(void* const* d_in, const int* in_sizes, int n_in,
                              void* d_out, int out_size, void* d_ws, size_t ws_size,
                              hipStream_t stream) {
  const float* x  = (const float*)d_in[0];
  const float* A  = (const float*)d_in[1];
  const float* B  = (const float*)d_in[2];
  const float* C  = (const float*)d_in[3];
  const float* D  = (const float*)d_in[4];
  const float* Wg = (const float*)d_in[5];
  const float* bh = (const float*)d_in[6];
  const float* by = (const float*)d_in[7];
  float* y = (float*)d_out;

  // Workspace layout (fp32): a_coef | b_coef | hs | Pc | Qc | Hst
  float* ws = (float*)d_ws;
  const size_t MD = (size_t)M_TOTAL * DIM;          // 16,777,216 elems each
  float* a_coef = ws;
  float* b_coef = a_coef + MD;
  float* hs     = b_coef + MD;
  float* Pc     = hs + MD;
  float* Qc     = Pc + (size_t)NCH * NCHUNK;
  float* Hst    = Qc + (size_t)NCH * NCHUNK;

  const dim3 blk(256);
  const dim3 gemm_grid(DIM / NT, M_TOTAL / MT);     // (8, 512)

  gate_gemm_kernel<<<gemm_grid, blk, 0, stream>>>(x, Wg, B, A, bh, a_coef, b_coef);
  scan_pass1<<<(NCH * NCHUNK) / 256, blk, 0, stream>>>(a_coef, b_coef, Pc, Qc);
  scan_pass2<<<NCH / 256, blk, 0, stream>>>(Pc, Qc, Hst);
  scan_pass3<<<(NCH * NCHUNK) / 256, blk, 0, stream>>>(a_coef, b_coef, Hst, hs);
  out_gemm_kernel<<<gemm_grid, blk, 0, stream>>>(hs, x, C, D, by, y);
}